// TV2D_12189117186125
// MI455X (gfx1250) — compile-verified
//
#include <hip/hip_runtime.h>
#include <cstdint>
#include <cmath>

// TV2D prox via FISTA on the dual (matches the JAX reference exactly):
//   100 iterations of: y = X - div(qv,qh); p_n = clip(q + TAU*D y, +/-lam);
//   q_n = p_n + b_k (p_n - p);  out = X - div(pv,ph)
// 2048x2048 f32. Working set ~84MB -> resident in MI455X's 192MB L2, so this
// is an L2-bandwidth-bound stencil. One fused kernel per iteration (kernel
// boundary supplies the global sync), float4-per-thread b128 traffic, padded
// 2048-stride dual buffers so boundary terms read zero padding.
// CDNA5 path: per-wave TENSOR_LOAD_TO_LDS (TDM) stages the tile-aligned
// pv/ph rows into LDS, overlapped with the y-stencil, completed with
// s_wait_tensorcnt before the momentum update.

#define TV_H 2048
#define TV_W 2048
#define TV_S 2048
#define TV_N ((size_t)TV_H * (size_t)TV_W)
#define TV_TAU 0.125f
#define TV_LAM 0.05f

#if defined(__has_builtin)
#if __has_builtin(__builtin_amdgcn_tensor_load_to_lds) && \
    __has_builtin(__builtin_amdgcn_s_wait_tensorcnt) && \
    __has_builtin(__builtin_amdgcn_readfirstlane)
#define TV_TDM 1
#endif
#endif

typedef unsigned int tv_u32x4 __attribute__((ext_vector_type(4)));
typedef int tv_i32x4 __attribute__((ext_vector_type(4)));
typedef int tv_i32x8 __attribute__((ext_vector_type(8)));

__device__ __forceinline__ float4 tv_ld4(const float* p) { return *(const float4*)p; }
__device__ __forceinline__ void tv_st4(float* p, float4 v) { *(float4*)p = v; }
__device__ __forceinline__ float tv_clamp(float v) {
    return fminf(fmaxf(v, -TV_LAM), TV_LAM);
}

#ifdef TV_TDM
// Issue one TDM load of a 128-float (512B) row into LDS for this wave.
// D# per CDNA5 ISA 8.3/8.4: group0 = {count=1 | lds_addr | global_addr | type=2},
// group1 = {data_size=4B, tensor_dim0/1=2048, tile_dim0=128, tile_dim1=1,
//           tensor_dim0_stride=2048 elements}. Groups 2/3 unused (<=2D tile).
// This toolchain's builtin takes 6 args: (u32x4, i32x8, i32x4, i32x4, i32x8, cpol).
__device__ __forceinline__ void tv_tdm_load_row(const float* gsrc, const float* ldst) {
    unsigned long long ga = (unsigned long long)(uintptr_t)gsrc;
    unsigned ga_lo = (unsigned)__builtin_amdgcn_readfirstlane((int)(unsigned)ga);
    unsigned ga_hi = (unsigned)__builtin_amdgcn_readfirstlane((int)(unsigned)(ga >> 32));
    // generic LDS pointer: low 32 bits are the LDS byte offset (aperture layout)
    unsigned loff  = (unsigned)__builtin_amdgcn_readfirstlane((int)(unsigned)(uintptr_t)ldst);
    tv_u32x4 g0;
    g0[0] = 1u;                                   // count=1, user descriptor
    g0[1] = loff;                                 // lds_addr (bytes)
    g0[2] = ga_lo;                                // global_addr[31:0]
    g0[3] = (ga_hi & 0x01FFFFFFu) | (2u << 30);   // global_addr[56:32] | type=2
    tv_i32x8 g1;
    g1[0] = (int)(2u << 16);      // workgroup_mask=0, data_size=2 (4 bytes)
    g1[1] = (int)(2048u << 16);   // tensor_dim0[15:0]=2048 in [31:16]
    g1[2] = (int)(2048u << 16);   // tensor_dim0[31:16]=0 | tensor_dim1[15:0]=2048
    g1[3] = (int)(128u << 16);    // tensor_dim1[31:16]=0 | tile_dim0=128
    g1[4] = 1;                    // tile_dim1=1, tile_dim2=0
    g1[5] = 2048;                 // tensor_dim0_stride[31:0] (elements)
    g1[6] = 0;                    // stride0[47:32]=0 | stride1[15:0]=0
    g1[7] = 0x40;                 // tensor_dim1_stride[47:16] (2048*2048 elems)
    tv_i32x4 gz4 = {0, 0, 0, 0};
    tv_i32x8 gz8 = {0, 0, 0, 0, 0, 0, 0, 0};
    __builtin_amdgcn_tensor_load_to_lds(g0, g1, gz4, gz4, gz8, 0);
}
#endif

// One FISTA iteration. Block = (32,8): each wave owns one row x 128 cols,
// each thread owns 4 consecutive cols (all b128 accesses 16B aligned).
// Dual buffers are stride-2048 with zero padding: qv/pv row H-1 == 0,
// qh/ph col W-1 == 0, so the y formula is branch-free except row 0 / col 0.
__global__ __launch_bounds__(256) void tv_iter_kernel(
    const float* __restrict__ X,
    float* __restrict__ pv, float* __restrict__ ph,
    float* __restrict__ qv, float* __restrict__ qh,
    float b)
{
#ifdef TV_TDM
    __shared__ float pvs[8][128];
    __shared__ float phs[8][128];
#endif
    const int tx = threadIdx.x;
    const int ty = threadIdx.y;
    const int i  = blockIdx.y * 8 + ty;
    const int j  = (blockIdx.x * 32 + tx) * 4;
    const size_t base = (size_t)i * TV_S + j;

#ifdef TV_TDM
    {   // async-stage this wave's pv/ph rows (tile-aligned, no halo needed)
        const size_t rowbase = (size_t)i * TV_S + (size_t)blockIdx.x * 128;
        tv_tdm_load_row(pv + rowbase, &pvs[ty][0]);
        tv_tdm_load_row(ph + rowbase, &phs[ty][0]);
    }
#endif

    const bool top  = (i == 0);
    const bool bot  = (i == TV_H - 1);
    const bool left = (j == 0);
    const int  ip   = bot ? i : (i + 1);
    const int  c4   = (j + 4 < TV_W) ? (j + 4) : (TV_W - 1);
    const size_t basep = (size_t)ip * TV_S + j;
    const size_t base4 = (size_t)i * TV_S + c4;

    // ---- gather stencil inputs (overlaps with the in-flight TDM) ----
    float4 Xi  = tv_ld4(X + base);
    float  x4  = X[base4];
    float4 Xp  = tv_ld4(X + basep);

    float4 qvi  = tv_ld4(qv + base);
    float  qvi4 = qv[base4];
    float4 qvm; float qvm4;
    if (top) { qvm = make_float4(0.f, 0.f, 0.f, 0.f); qvm4 = 0.f; }
    else     { qvm = tv_ld4(qv + base - TV_S); qvm4 = qv[base4 - TV_S]; }
    float4 qvp = tv_ld4(qv + basep);          // row H-1 is zero padding

    float4 qhi  = tv_ld4(qh + base);
    float  qhil = left ? 0.f : qh[base - 1];
    float  qhi4 = qh[base4];                  // col W-1 is zero padding
    float4 qhp  = tv_ld4(qh + basep);
    float  qhpl = left ? 0.f : qh[basep - 1];

    // y[i,c] = X - qv[i-1,c] + qv[i,c] - qh[i,c-1] + qh[i,c]
    float y0 = Xi.x - qvm.x + qvi.x - qhil  + qhi.x;
    float y1 = Xi.y - qvm.y + qvi.y - qhi.x + qhi.y;
    float y2 = Xi.z - qvm.z + qvi.z - qhi.y + qhi.z;
    float y3 = Xi.w - qvm.w + qvi.w - qhi.z + qhi.w;
    float y4 = x4   - qvm4  + qvi4  - qhi.w + qhi4;

    // y[i+1,c]
    float z0 = Xp.x - qvi.x + qvp.x - qhpl  + qhp.x;
    float z1 = Xp.y - qvi.y + qvp.y - qhp.x + qhp.y;
    float z2 = Xp.z - qvi.z + qvp.z - qhp.y + qhp.z;
    float z3 = Xp.w - qvi.w + qvp.w - qhp.z + qhp.w;

    // ---- momentum inputs: complete the TDM and read from LDS ----
    float4 pvo, pho;
#ifdef TV_TDM
    __builtin_amdgcn_s_wait_tensorcnt(0);
    pvo = *(const float4*)&pvs[ty][tx * 4];
    pho = *(const float4*)&phs[ty][tx * 4];
#else
    pvo = tv_ld4(pv + base);
    pho = tv_ld4(ph + base);
#endif

    float4 pvn, phn;
    pvn.x = tv_clamp(qvi.x + TV_TAU * (z0 - y0));
    pvn.y = tv_clamp(qvi.y + TV_TAU * (z1 - y1));
    pvn.z = tv_clamp(qvi.z + TV_TAU * (z2 - y2));
    pvn.w = tv_clamp(qvi.w + TV_TAU * (z3 - y3));

    phn.x = tv_clamp(qhi.x + TV_TAU * (y1 - y0));
    phn.y = tv_clamp(qhi.y + TV_TAU * (y2 - y1));
    phn.z = tv_clamp(qhi.z + TV_TAU * (y3 - y2));
    phn.w = tv_clamp(qhi.w + TV_TAU * (y4 - y3));
    if (j + 4 >= TV_W) phn.w = 0.f;  // keep col W-1 padding at zero

    float4 qvn, qhn;
    qvn.x = pvn.x + b * (pvn.x - pvo.x);
    qvn.y = pvn.y + b * (pvn.y - pvo.y);
    qvn.z = pvn.z + b * (pvn.z - pvo.z);
    qvn.w = pvn.w + b * (pvn.w - pvo.w);
    qhn.x = phn.x + b * (phn.x - pho.x);
    qhn.y = phn.y + b * (phn.y - pho.y);
    qhn.z = phn.z + b * (phn.z - pho.z);
    qhn.w = phn.w + b * (phn.w - pho.w);

    if (!bot) {                       // keep row H-1 padding at zero
        tv_st4(pv + base, pvn);
        tv_st4(qv + base, qvn);
    }
    tv_st4(ph + base, phn);
    tv_st4(qh + base, qhn);
}

// out = X - pv[i-1] + pv[i] - ph[i,j-1] + ph[i,j]  (padding supplies zeros)
__global__ __launch_bounds__(256) void tv_final_kernel(
    const float* __restrict__ X,
    const float* __restrict__ pv, const float* __restrict__ ph,
    float* __restrict__ out)
{
    const int i = blockIdx.y * 8 + threadIdx.y;
    const int j = (blockIdx.x * 32 + threadIdx.x) * 4;
    const size_t base = (size_t)i * TV_S + j;
    float4 x  = tv_ld4(X + base);
    float4 a  = tv_ld4(pv + base);
    float4 am = (i == 0) ? make_float4(0.f, 0.f, 0.f, 0.f) : tv_ld4(pv + base - TV_S);
    float4 hc = tv_ld4(ph + base);
    float  hl = (j == 0) ? 0.f : ph[base - 1];
    float4 r;
    r.x = x.x - am.x + a.x - hl   + hc.x;
    r.y = x.y - am.y + a.y - hc.x + hc.y;
    r.z = x.z - am.z + a.z - hc.y + hc.z;
    r.w = x.w - am.w + a.w - hc.z + hc.w;
    tv_st4(out + base, r);
}

__global__ void tv_zero_kernel(float4* p, size_t n4) {
    size_t idx = (size_t)blockIdx.x * blockDim.x + threadIdx.x;
    size_t stride = (size_t)gridDim.x * blockDim.x;
    const float4 z = make_float4(0.f, 0.f, 0.f, 0.f);
    for (; idx < n4; idx += stride) p[idx] = z;
}

extern "C" void kernel_launch(void* const* d_in, const int* in_sizes, int n_in,
                              void* d_out, int out_size, void* d_ws, size_t ws_size,
                              hipStream_t stream) {
    (void)in_sizes; (void)n_in; (void)out_size; (void)ws_size;
    const float* X = (const float*)d_in[0];
    float* out = (float*)d_out;
    float* pv = (float*)d_ws;        // [2048][2048], rows 0..2046 live, row 2047 = 0 pad
    float* ph = pv + TV_N;           // [2048][2048], cols 0..2046 live, col 2047 = 0 pad
    float* qv = ph + TV_N;
    float* qh = out;                 // reuse d_out as qh scratch; final pass overwrites it

    // zero-init duals (padding invariants depend on it; done every call)
    tv_zero_kernel<<<2048, 256, 0, stream>>>((float4*)d_ws, (3 * TV_N) / 4);
    tv_zero_kernel<<<1024, 256, 0, stream>>>((float4*)d_out, TV_N / 4);

    dim3 grid(TV_W / 128, TV_H / 8);
    dim3 blk(32, 8);
    float t = 1.0f;
    for (int k = 0; k < 100; ++k) {   // MAX_ITER = 100
        float tn = 0.5f * (1.0f + sqrtf(1.0f + 4.0f * t * t));
        float b = (t - 1.0f) / tn;    // b_0 = 0, matching the reference
        t = tn;
        tv_iter_kernel<<<grid, blk, 0, stream>>>(X, pv, ph, qv, qh, b);
    }
    tv_final_kernel<<<grid, blk, 0, stream>>>(X, pv, ph, out);
}